// NeuralODE_77644418777195
// MI455X (gfx1250) — compile-verified
//
#include <hip/hip_runtime.h>
#include <math.h>

typedef __attribute__((ext_vector_type(2))) float v2f;
typedef __attribute__((ext_vector_type(8))) float v8f;

#define DATA    64
#define WIDTH   256
#define TPTS    16
#define NINT    (TPTS - 1)
#define MAXSUB  8
#define MT      16      // batch rows per workgroup (one WMMA M-tile)
#define NTHR    512     // 16 waves of 32
#define BATCH   512

#define RTOLC   1e-3f
#define ATOLC   1e-6f
#define SAFETY  0.9f
#define FMINC   0.2f
#define FMAXC   10.0f

// Tsit5: row s holds coefficients of k1..k6 for the stage-s input state.
// Row 6 is the b-row (y_new = y + dt * b.k), whose f-eval is k7 (FSAL stage).
__device__ __constant__ float STAGE_A[7][6] = {
  { 0.f, 0.f, 0.f, 0.f, 0.f, 0.f },
  { 0.161f, 0.f, 0.f, 0.f, 0.f, 0.f },
  { -0.008480655492356989f, 0.335480655492357f, 0.f, 0.f, 0.f, 0.f },
  { 2.8971530571054935f, -6.359448489975075f, 4.3622954328695815f, 0.f, 0.f, 0.f },
  { 5.325864828439257f, -11.748883564062828f, 7.4955393428898365f, -0.09249506636175525f, 0.f, 0.f },
  { 5.86145544294642f, -12.92096931784711f, 8.159367898576159f, -0.071584973281401f, -0.028269050394068383f, 0.f },
  { 0.09646076681806523f, 0.01f, 0.4798896504144996f, 1.379008574103742f, -3.290069515436081f, 2.324710524099774f },
};
__device__ __constant__ float ECOEF[7] = {
  -0.001780011052226f, -0.000816434459657f, 0.007880878010262f, -0.144711007173263f,
   0.582357165452555f, -0.458082105929187f, 0.015151515151515152f };

// Softplus on raw hardware transcendentals (v_exp_f32 = 2^x, v_log_f32 = log2 x):
//   softplus(x) = max(x,0) + ln2 * log2(1 + 2^(-|x| * log2e))
// log2 argument is in (1,2] -> no denorm/inf guards needed; ~1 ulp accuracy.
__device__ __forceinline__ float softplusf(float x) {
  const float LOG2E = 1.4426950408889634f;
  const float LN2   = 0.6931471805599453f;
  float e = __builtin_amdgcn_exp2f(-fabsf(x) * LOG2E);
  return fmaxf(x, 0.0f) + LN2 * __builtin_amdgcn_logf(1.0f + e);
}

__global__ __launch_bounds__(NTHR, 1)
void node_tsit5_kernel(const float* __restrict__ ts, const float* __restrict__ y0,
                       const float* __restrict__ W0, const float* __restrict__ b0,
                       const float* __restrict__ W1, const float* __restrict__ b1,
                       const float* __restrict__ W2, const float* __restrict__ b2,
                       float* __restrict__ out)
{
  __shared__ __align__(16) float sY [MT][DATA];      // current state
  __shared__ __align__(16) float sYn[MT][DATA];      // proposed state
  __shared__ __align__(16) float sZ [MT][DATA];      // stage input
  __shared__ __align__(16) float sK [7][MT][DATA];   // k1..k7
  __shared__ __align__(16) float sH0[MT][WIDTH];     // hidden 1
  __shared__ __align__(16) float sH1[MT][WIDTH];     // hidden 2
  __shared__ __align__(16) float sFp[4][MT][DATA];   // split-K partials (GEMM3)
  __shared__ float sT[MT], sDt[MT], sDtc[MT], sStep[MT], sDone[MT];
  __shared__ float sTs[TPTS];

  const int tid  = threadIdx.x;
  const int lane = tid & 31;
  const int wav  = tid >> 5;       // 0..15
  const int kh   = lane >> 4;      // K-half selector inside WMMA fragments
  const int l16  = lane & 15;      // M (A-frag) / N (B,C,D frags) index
  const int row0 = blockIdx.x * MT;
  const int er   = wav;            // elementwise: one wave per batch row
  const int ec   = lane * 2;       // elementwise: 2 dims per lane (64 = 32*2)

  // ---- init: load y0, emit snapshot at t=ts[0] ----
  if (tid < TPTS) sTs[tid] = ts[tid];
  {
    const float* yp = y0 + (row0 + er) * DATA + ec;
    float a0 = yp[0], a1 = yp[1];
    sY[er][ec] = a0; sY[er][ec + 1] = a1;
    float* op = out + ((size_t)(row0 + er) * TPTS + 0) * DATA + ec;
    op[0] = a0; op[1] = a1;
  }
  if (lane == 0) sDt[er] = ts[1] - ts[0];
  __syncthreads();

  // One MLP evaluation of f(sZ) -> dst[MT][DATA]. Assumes entry after a barrier.
  auto mlp = [&](float (*dst)[DATA]) {
    // --- GEMM1: sH0 = softplus(sZ[16x64] * W0^T + b0), wave -> one 16-col N-tile
    {
      const int n = wav * 16 + l16;
      v8f acc = {};
      const float* wp = W0 + n * DATA + kh * 2;
      #pragma unroll
      for (int kk = 0; kk < DATA / 4; ++kk) {
        const int k = kk * 4 + kh * 2;
        v2f a = *(const v2f*)(&sZ[l16][k]);            // A[m=l16][k], A[m][k+1]
        v2f b = *(const v2f*)(wp + kk * 4);            // B[k][n]=W0[n][k], B[k+1][n]
        acc = __builtin_amdgcn_wmma_f32_16x16x4_f32(false, a, false, b,
                                                    (short)0, acc, false, false);
      }
      const float bias = b0[n];
      #pragma unroll
      for (int v = 0; v < 8; ++v)
        sH0[v + kh * 8][n] = softplusf(acc[v] + bias);
    }
    __syncthreads();
    // --- GEMM2: sH1 = softplus(sH0[16x256] * W1^T + b1)
    {
      const int n = wav * 16 + l16;
      v8f acc = {};
      const float* wp = W1 + n * WIDTH + kh * 2;
      #pragma unroll 8
      for (int kk = 0; kk < WIDTH / 4; ++kk) {
        const int k = kk * 4 + kh * 2;
        v2f a = *(const v2f*)(&sH0[l16][k]);
        v2f b = *(const v2f*)(wp + kk * 4);
        acc = __builtin_amdgcn_wmma_f32_16x16x4_f32(false, a, false, b,
                                                    (short)0, acc, false, false);
      }
      const float bias = b1[n];
      #pragma unroll
      for (int v = 0; v < 8; ++v)
        sH1[v + kh * 8][n] = softplusf(acc[v] + bias);
    }
    __syncthreads();
    // --- GEMM3: F = sH1[16x256] * W2^T, 4 N-tiles x 4-way split-K over 16 waves
    {
      const int dtile = wav & 3;
      const int kq    = wav >> 2;
      const int d     = dtile * 16 + l16;
      v8f acc = {};
      const float* wp = W2 + d * WIDTH + kq * 64 + kh * 2;
      #pragma unroll
      for (int kk = 0; kk < 16; ++kk) {
        const int k = kq * 64 + kk * 4 + kh * 2;
        v2f a = *(const v2f*)(&sH1[l16][k]);
        v2f b = *(const v2f*)(wp + kk * 4);
        acc = __builtin_amdgcn_wmma_f32_16x16x4_f32(false, a, false, b,
                                                    (short)0, acc, false, false);
      }
      #pragma unroll
      for (int v = 0; v < 8; ++v)
        sFp[kq][v + kh * 8][d] = acc[v];
    }
    __syncthreads();
    // --- combine split-K partials + b2
    #pragma unroll
    for (int j = 0; j < 2; ++j) {
      const int c = ec + j;
      dst[er][c] = sFp[0][er][c] + sFp[1][er][c] + sFp[2][er][c] + sFp[3][er][c] + b2[c];
    }
    __syncthreads();
  };

  #pragma unroll 1
  for (int iv = 0; iv < NINT; ++iv) {
    const float t1 = sTs[iv + 1];
    if (lane == 0) sT[er] = sTs[iv];
    __syncthreads();

    #pragma unroll 1
    for (int ss = 0; ss < MAXSUB; ++ss) {
      // --- substep setup (one lane per row) ---
      if (lane == 0) {
        float t = sT[er], dt = sDt[er];
        float rem  = t1 - t;
        bool  done = rem <= 1e-6f;
        sDtc[er]  = done ? dt : fminf(dt, rem);
        sDone[er] = done ? 1.0f : 0.0f;
      }
      __syncthreads();
      const float dtc = sDtc[er];

      // --- 7 stages: build stage state, evaluate MLP ---
      #pragma unroll 1
      for (int s = 0; s < 7; ++s) {
        #pragma unroll
        for (int j = 0; j < 2; ++j) {
          const int c = ec + j;
          float acc = 0.0f;
          for (int q = 0; q < s && q < 6; ++q)           // q<s avoids stale-k NaN*0
            acc = fmaf(STAGE_A[s][q], sK[q][er][c], acc);
          float z = fmaf(dtc, acc, sY[er][c]);
          sZ[er][c] = z;
          if (s == 6) sYn[er][c] = z;                    // stage 7 input IS y_new
        }
        __syncthreads();
        mlp(sK[s]);
      }

      // --- embedded error + I-controller ---
      float esq = 0.0f;
      #pragma unroll
      for (int j = 0; j < 2; ++j) {
        const int c = ec + j;
        float e = 0.0f;
        #pragma unroll
        for (int q = 0; q < 7; ++q) e = fmaf(ECOEF[q], sK[q][er][c], e);
        e *= dtc;
        float scale = fmaf(RTOLC, fmaxf(fabsf(sY[er][c]), fabsf(sYn[er][c])), ATOLC);
        float r = e / scale;
        esq = fmaf(r, r, esq);
      }
      #pragma unroll
      for (int o = 16; o > 0; o >>= 1) esq += __shfl_xor(esq, o, 32);   // wave32
      if (lane == 0) {
        float errn   = sqrtf(esq * (1.0f / DATA));
        bool  accept = errn <= 1.0f;
        float factor = SAFETY * __powf(fmaxf(errn, 1e-10f), -0.2f);
        factor = fminf(fmaxf(factor, FMINC), FMAXC);
        bool  done = sDone[er] != 0.0f;
        bool  step = accept && !done;
        if (step)  sT[er] += dtc;
        if (!done) sDt[er] = dtc * factor;
        sStep[er] = step ? 1.0f : 0.0f;
      }
      __syncthreads();
      if (sStep[er] != 0.0f) {
        sY[er][ec]     = sYn[er][ec];
        sY[er][ec + 1] = sYn[er][ec + 1];
      }
      __syncthreads();
    }

    // emit snapshot for t = ts[iv+1]
    float* op = out + ((size_t)(row0 + er) * TPTS + (iv + 1)) * DATA + ec;
    op[0] = sY[er][ec];
    op[1] = sY[er][ec + 1];
    __syncthreads();
  }
}

extern "C" void kernel_launch(void* const* d_in, const int* in_sizes, int n_in,
                              void* d_out, int out_size, void* d_ws, size_t ws_size,
                              hipStream_t stream) {
  (void)in_sizes; (void)n_in; (void)out_size; (void)d_ws; (void)ws_size;
  const float* ts = (const float*)d_in[0];
  const float* y0 = (const float*)d_in[1];
  const float* W0 = (const float*)d_in[2];
  const float* b0 = (const float*)d_in[3];
  const float* W1 = (const float*)d_in[4];
  const float* b1 = (const float*)d_in[5];
  const float* W2 = (const float*)d_in[6];
  const float* b2 = (const float*)d_in[7];
  float* out = (float*)d_out;
  node_tsit5_kernel<<<dim3(BATCH / MT), dim3(NTHR), 0, stream>>>(
      ts, y0, W0, b0, W1, b1, W2, b2, out);
}